// Sampler_61615600828711
// MI455X (gfx1250) — compile-verified
//
#include <hip/hip_runtime.h>
#include <hip/hip_bf16.h>
#include <stdint.h>

// ---------------------------------------------------------------------------
// Fused top-k/top-p sampling for B=128 rows, V=128000 vocab, k=20.
// Single streaming pass over the logits (65.5 MB -> ~2.8us HBM floor on
// MI455X @ 23.3 TB/s) using the CDNA5 async global->LDS data mover
// (global_load_async_to_lds_b128 / s_wait_asynccnt) as a depth-16 prefetch
// pipeline (128 KB LDS in flight per WG ~= per-block BW share x latency).
// Per-thread branchless online softmax + register-resident sorted top-20,
// then wave/block argmax merges, then a 20-element CDF walk.
// ---------------------------------------------------------------------------

#define VOCAB   128000
#define TPB     512
#define NWAVE   16          // 512 / wave32
#define KMAX    20
#define CHUNK   2048        // floats per chunk = TPB * 4
#define NCHUNK  62          // 62 * 2048 = 126976 streamed via async
#define TAILBASE (NCHUNK * CHUNK)   // 126976; remaining 1024 via plain loads
#define DEPTH   16          // async chunks in flight (128 KB LDS staging)

#define STR_(x) #x
#define WAIT_ASYNC_LE(n) asm volatile("s_wait_asynccnt " STR_(n) ::: "memory")
#define WAIT_DS0()       asm volatile("s_wait_dscnt 0" ::: "memory")

// Issue one 16B-per-lane async copy global -> LDS. Each lane supplies its own
// LDS byte offset (low 32 bits of the generic LDS pointer) and 64-bit global
// address; completion is tracked with ASYNCcnt (in-order per wave).
__device__ __forceinline__ void async_issue(const float* g, const float* l) {
    uint32_t lds32 = (uint32_t)(uintptr_t)l;          // LDS byte offset
    uint64_t ga    = (uint64_t)(uintptr_t)g;          // global address (VGPR pair)
    asm volatile("global_load_async_to_lds_b128 %0, %1, off"
                 :: "v"(lds32), "v"(ga) : "memory");
}

// Sorted-descending top-20 insert, fully unrolled (register resident).
__device__ __forceinline__ void ins20(float v, int idx, float tv[KMAX], int ti[KMAX]) {
    if (v > tv[KMAX - 1]) {
        float cv = v; int ci = idx;
#pragma unroll
        for (int j = 0; j < KMAX; ++j) {
            if (cv > tv[j]) {
                float t0 = tv[j]; tv[j] = cv; cv = t0;
                int   t1 = ti[j]; ti[j] = ci; ci = t1;
            }
        }
    }
}

// Branchless online (max, sum-exp): exactly one v_exp_f32 per element.
__device__ __forceinline__ void onl(float v, float& rm, float& rs) {
    float d  = v - rm;
    float e  = __expf(-fabsf(d));     // exp(-|v-rm|)
    bool  gt = d > 0.0f;
    rs = gt ? (rs * e + 1.0f) : (rs + e);
    rm = gt ? v : rm;
}

// Static-index head extraction from a sorted register list (cmov chain).
__device__ __forceinline__ float selv(const float tv[KMAX], int hp) {
    float h = tv[0];
#pragma unroll
    for (int j = 1; j < KMAX; ++j) h = (hp == j) ? tv[j] : h;
    return (hp < KMAX) ? h : -INFINITY;
}
__device__ __forceinline__ int seli(const int ti[KMAX], int hp) {
    int h = ti[0];
#pragma unroll
    for (int j = 1; j < KMAX; ++j) h = (hp == j) ? ti[j] : h;
    return h;
}

__global__ __launch_bounds__(TPB)
void Sampler_61615600828711_kernel(const float* __restrict__ logits,
                                   const float* __restrict__ temp,
                                   const float* __restrict__ topp,
                                   const float* __restrict__ uarr,
                                   const int*   __restrict__ topk,
                                   int*         __restrict__ out) {
    __shared__ __align__(16) float stage[DEPTH][CHUNK];   // 128 KB async staging
    __shared__ float wTopV[NWAVE * KMAX];
    __shared__ int   wTopI[NWAVE * KMAX];
    __shared__ float fV[KMAX];
    __shared__ int   fI[KMAX];
    __shared__ float red[NWAVE];
    __shared__ float sM, sZ;

    const int tid  = threadIdx.x;
    const int lane = tid & 31;
    const int w    = tid >> 5;
    const int row  = blockIdx.x;

    const float invT = 1.0f / temp[0];
    const float* rowPtr = logits + (size_t)row * VOCAB;

    float tv[KMAX]; int ti[KMAX];
#pragma unroll
    for (int j = 0; j < KMAX; ++j) { tv[j] = -INFINITY; ti[j] = 0; }
    float rm = -INFINITY, rs = 0.0f;

    // ---- prime the async pipeline (DEPTH chunks in flight) ----
#pragma unroll
    for (int d = 0; d < DEPTH; ++d)
        async_issue(rowPtr + d * CHUNK + tid * 4, &stage[d][tid * 4]);

    // ---- streaming loop: each lane consumes only the 16B it fetched ----
    for (int c = 0; c < NCHUNK; ++c) {
        if (c < NCHUNK - DEPTH) { WAIT_ASYNC_LE(15); }  // oldest of 16 done
        else                    { WAIT_ASYNC_LE(0);  }  // one-time drain at tail
        const int b = c & (DEPTH - 1);
        float4 x = *reinterpret_cast<const float4*>(&stage[b][tid * 4]);
        WAIT_DS0();                                     // regs hold data; buffer reusable
        if (c + DEPTH < NCHUNK)
            async_issue(rowPtr + (c + DEPTH) * CHUNK + tid * 4, &stage[b][tid * 4]);

        const int base = c * CHUNK + tid * 4;
        float v0 = x.x * invT, v1 = x.y * invT, v2 = x.z * invT, v3 = x.w * invT;
        onl(v0, rm, rs); ins20(v0, base + 0, tv, ti);
        onl(v1, rm, rs); ins20(v1, base + 1, tv, ti);
        onl(v2, rm, rs); ins20(v2, base + 2, tv, ti);
        onl(v3, rm, rs); ins20(v3, base + 3, tv, ti);
    }
    // ---- tail: 1024 elements, 2 per thread, plain global loads ----
    {
        const int base = TAILBASE + tid * 2;
        float a0 = rowPtr[base + 0] * invT;
        float a1 = rowPtr[base + 1] * invT;
        onl(a0, rm, rs); ins20(a0, base + 0, tv, ti);
        onl(a1, rm, rs); ins20(a1, base + 1, tv, ti);
    }

    // ---- wave-level 32-way merge: 20 rounds of wave-argmax over list heads ----
    {
        int hp = 0;
        for (int r = 0; r < KMAX; ++r) {
            float hv = selv(tv, hp);
            int   hi = seli(ti, hp);
            float bv = hv; int bl = lane;
#pragma unroll
            for (int m = 16; m; m >>= 1) {
                float ov = __shfl_xor(bv, m, 32);
                int   ol = __shfl_xor(bl, m, 32);
                if (ov > bv || (ov == bv && ol < bl)) { bv = ov; bl = ol; }
            }
            int wi = __shfl(hi, bl, 32);
            if (lane == 0) { wTopV[w * KMAX + r] = bv; wTopI[w * KMAX + r] = wi; }
            if (lane == bl) hp++;
        }
    }
    __syncthreads();

    // ---- block (M, Z) reduction ----
    float bm = rm;
#pragma unroll
    for (int m = 16; m; m >>= 1) bm = fmaxf(bm, __shfl_xor(bm, m, 32));
    if (lane == 0) red[w] = bm;
    __syncthreads();
    if (tid == 0) {
        float mm = red[0];
        for (int i = 1; i < NWAVE; ++i) mm = fmaxf(mm, red[i]);
        sM = mm;
    }
    __syncthreads();
    const float M = sM;
    float contrib = rs * __expf(rm - M);
#pragma unroll
    for (int m = 16; m; m >>= 1) contrib += __shfl_xor(contrib, m, 32);
    if (lane == 0) red[w] = contrib;
    __syncthreads();
    if (tid == 0) {
        float z = 0.0f;
        for (int i = 0; i < NWAVE; ++i) z += red[i];
        sZ = z;
    }
    __syncthreads();

    // ---- final merge: wave 0 extracts global top-20 from 320 candidates ----
    for (int r = 0; r < KMAX; ++r) {
        if (w == 0) {
            float bv = -INFINITY; int bp = 0;
#pragma unroll
            for (int j = 0; j < (NWAVE * KMAX) / 32; ++j) {   // 10 per lane
                int pos = lane + (j << 5);
                float val = wTopV[pos];
                if (val > bv || (val == bv && pos < bp)) { bv = val; bp = pos; }
            }
#pragma unroll
            for (int m = 16; m; m >>= 1) {
                float ov = __shfl_xor(bv, m, 32);
                int   op = __shfl_xor(bp, m, 32);
                if (ov > bv || (ov == bv && op < bp)) { bv = ov; bp = op; }
            }
            if (lane == 0) { fV[r] = bv; fI[r] = wTopI[bp]; wTopV[bp] = -INFINITY; }
        }
        __syncthreads();
    }

    // ---- top-p prefix mask + inverse-CDF pick (thread 0, 20 elements) ----
    if (tid == 0) {
        const float Z  = sZ;
        const float uu = uarr[row];
        const float tp = topp[0];
        int k = topk[0]; if (k > KMAX) k = KMAX;

        float pv[KMAX]; int kept[KMAX];
        float S = 0.0f, cb = 0.0f;
#pragma unroll
        for (int i = 0; i < KMAX; ++i) {
            float pi = expf(fV[i] - M) / Z;
            int   kp = (i < k) && (cb < tp);
            pv[i] = pi; kept[i] = kp;
            if (kp) S += pi;
            cb += pi;                       // cumulative over ALL sorted probs
        }
        const float thr = uu * S;           // cdf < u  <=>  kept-cumsum < u*S
        float cacc = 0.0f; int idx = 0;
#pragma unroll
        for (int i = 0; i < KMAX; ++i) {
            if (kept[i]) { cacc += pv[i]; if (cacc < thr) idx++; }
        }
        if (idx > KMAX - 1) idx = KMAX - 1;
        out[row] = fI[idx];
    }
}

extern "C" void kernel_launch(void* const* d_in, const int* in_sizes, int n_in,
                              void* d_out, int out_size, void* d_ws, size_t ws_size,
                              hipStream_t stream) {
    (void)d_ws; (void)ws_size; (void)n_in;
    const float* logits = (const float*)d_in[0];
    const float* temp   = (const float*)d_in[1];
    const float* topp   = (const float*)d_in[2];
    const float* u      = (const float*)d_in[3];
    const int*   topk   = (const int*)d_in[4];
    int* out = (int*)d_out;
    const int B = in_sizes[3];      // u has one entry per row
    (void)out_size;
    Sampler_61615600828711_kernel<<<B, TPB, 0, stream>>>(logits, temp, topp, u, topk, out);
}